// Cosear_Stat_8358006358397
// MI455X (gfx1250) — compile-verified
//
#include <hip/hip_runtime.h>
#include <stdint.h>

// ---------------------------------------------------------------------------
// Windowed empirical CDF via exact fine-binned histogram + bucket partition.
//   bins  : 2^20 uniform bins over [-8, 8]  (monotone map, clamped edges)
//   counts: uint32 atomics (deterministic)
//   sums  : int64 fixed-point (value * 2^24) atomics (deterministic)
// Query: exact prefix at bin + exact float4-vectorized scan of the single
// boundary bin (index-masked edges). Workspace ~156 MB.
// gfx1250 paths: global_load_async_to_lds_b128 staging + s_wait_asynccnt.
// ---------------------------------------------------------------------------

#define NBINS      (1 << 20)
#define CHUNK      4096
#define NCHUNK     (NBINS / CHUNK)      // 256
#define RLO        (-8.0f)
#define RSCALE     (65536.0f)           // NBINS / 16
#define FIXSCALE   16777216.0f          // 2^24
#define INV_FIX    (1.0 / 16777216.0)
#define DELTA_W    0.05f
#define TILE_F     4096                 // 16 KB LDS staging tile
#define HIST_THREADS 256

__device__ __forceinline__ int bin_of(float x) {
  // monotone non-decreasing in x (IEEE sub/mul monotone); clamped edge bins
  // stay exact because boundary bins are always scanned with exact compares.
  float t = (x - RLO) * RSCALE;
  t = fmaxf(t, 0.0f);                  // NaN -> 0 (consistent both passes)
  t = fminf(t, (float)(NBINS - 1));
  return (int)t;
}

__device__ __forceinline__ long long fixp(float x) {
  return __float2ll_rn(x * FIXSCALE);  // identical in hist & query passes
}

// --------------------------- pass 0: zero tables ---------------------------
__global__ __launch_bounds__(256)
void k_init(unsigned* __restrict__ binCnt, unsigned* __restrict__ binFill,
            long long* __restrict__ binSum) {
  int i = blockIdx.x * blockDim.x + threadIdx.x;
  int stride = gridDim.x * blockDim.x;
  for (; i < NBINS; i += stride) { binCnt[i] = 0u; binFill[i] = 0u; binSum[i] = 0ll; }
}

// ------------- pass 1: histogram, staged through LDS w/ async DMA ----------
__global__ __launch_bounds__(HIST_THREADS)
void k_hist(const float* __restrict__ m, int nm,
            unsigned* __restrict__ binCnt,
            unsigned long long* __restrict__ binSum) {
  __shared__ __align__(16) float tile[TILE_F];
  const int tid = threadIdx.x;
  const int ntile = (nm + TILE_F - 1) / TILE_F;
  for (int t = blockIdx.x; t < ntile; t += gridDim.x) {
    const int base = t * TILE_F;
    const int n = min(TILE_F, nm - base);
    if (n == TILE_F) {
      // gfx1250 async global->LDS copy: 4 x b128 per thread covers 16 KB.
      // LDS flat aperture keeps the LDS byte offset in addr[31:0] (ISA 10.2).
      #pragma unroll
      for (int c = 0; c < TILE_F; c += HIST_THREADS * 4) {
        unsigned lds_addr = (unsigned)(uintptr_t)(&tile[c + tid * 4]);
        unsigned long long gaddr =
            (unsigned long long)(uintptr_t)(m + base + c + tid * 4);
        asm volatile("global_load_async_to_lds_b128 %0, %1, off"
                     :: "v"(lds_addr), "v"(gaddr) : "memory");
      }
      asm volatile("s_wait_asynccnt 0x0" ::: "memory");
      __syncthreads();
    } else {
      for (int i = tid; i < n; i += HIST_THREADS) tile[i] = m[base + i];
      __syncthreads();
    }
    for (int i = tid; i < n; i += HIST_THREADS) {
      float x = tile[i];
      int b = bin_of(x);
      atomicAdd(&binCnt[b], 1u);
      atomicAdd(&binSum[b], (unsigned long long)fixp(x));  // 2's-complement add
    }
    __syncthreads();  // protect tile before next async round
  }
}

// --------------------- pass 2a: per-chunk totals (reduce) ------------------
__global__ __launch_bounds__(256)
void k_reduce(const unsigned* __restrict__ binCnt,
              const long long* __restrict__ binSum,
              unsigned* __restrict__ chunkCnt,
              long long* __restrict__ chunkSum) {
  __shared__ unsigned sc[256];
  __shared__ long long ss[256];
  const int tid = threadIdx.x, blk = blockIdx.x;
  unsigned c = 0; long long s = 0;
  for (int i = tid; i < CHUNK; i += 256) {
    int b = blk * CHUNK + i;
    c += binCnt[b]; s += binSum[b];
  }
  sc[tid] = c; ss[tid] = s; __syncthreads();
  for (int off = 128; off > 0; off >>= 1) {
    if (tid < off) { sc[tid] += sc[tid + off]; ss[tid] += ss[tid + off]; }
    __syncthreads();
  }
  if (tid == 0) { chunkCnt[blk] = sc[0]; chunkSum[blk] = ss[0]; }
}

// ---------------- pass 2b: scan the 256 chunk totals (tiny) ----------------
__global__ void k_scanchunks(const unsigned* __restrict__ chunkCnt,
                             const long long* __restrict__ chunkSum,
                             unsigned* __restrict__ chunkOffCnt,
                             long long* __restrict__ chunkOffSum,
                             unsigned* __restrict__ binStart,
                             long long* __restrict__ cumSum) {
  if (threadIdx.x == 0 && blockIdx.x == 0) {
    unsigned rc = 0; long long rs = 0;
    for (int i = 0; i < NCHUNK; ++i) {
      chunkOffCnt[i] = rc; chunkOffSum[i] = rs;
      rc += chunkCnt[i];   rs += chunkSum[i];
    }
    binStart[NBINS] = rc;  // total element count
    cumSum[NBINS]   = rs;  // total fixed-point sum
  }
}

// ------------- pass 2c: exclusive scan within each 4096-bin chunk ----------
__global__ __launch_bounds__(256)
void k_scanwithin(const unsigned* __restrict__ binCnt,
                  const long long* __restrict__ binSum,
                  const unsigned* __restrict__ chunkOffCnt,
                  const long long* __restrict__ chunkOffSum,
                  unsigned* __restrict__ binStart,
                  long long* __restrict__ cumSum) {
  __shared__ unsigned sc[256];
  __shared__ long long ss[256];
  const int tid = threadIdx.x, blk = blockIdx.x;
  unsigned carryC = chunkOffCnt[blk];
  long long carryS = chunkOffSum[blk];
  for (int r = 0; r < CHUNK; r += 256) {
    const int b = blk * CHUNK + r + tid;
    const unsigned vc = binCnt[b];
    const long long vs = binSum[b];
    sc[tid] = vc; ss[tid] = vs; __syncthreads();
    for (int off = 1; off < 256; off <<= 1) {       // Hillis-Steele inclusive
      unsigned tc = (tid >= off) ? sc[tid - off] : 0u;
      long long ts = (tid >= off) ? ss[tid - off] : 0ll;
      __syncthreads();
      sc[tid] += tc; ss[tid] += ts;
      __syncthreads();
    }
    binStart[b] = carryC + (sc[tid] - vc);          // exclusive prefix
    cumSum[b]   = carryS + (ss[tid] - vs);
    carryC += sc[255]; carryS += ss[255];
    __syncthreads();                                // before LDS reuse
  }
}

// ------------------- pass 3: scatter values into buckets -------------------
__global__ __launch_bounds__(256)
void k_scatter(const float* __restrict__ m, int nm,
               const unsigned* __restrict__ binStart,
               unsigned* __restrict__ binFill,
               float* __restrict__ bucketVals) {
  int i4 = (blockIdx.x * blockDim.x + threadIdx.x) * 4;
  const int stride = gridDim.x * blockDim.x * 4;
  for (; i4 + 3 < nm; i4 += stride) {
    const float4 v = *(const float4*)(m + i4);
    const float xs[4] = {v.x, v.y, v.z, v.w};
    #pragma unroll
    for (int k = 0; k < 4; ++k) {
      const float x = xs[k];
      const int b = bin_of(x);
      const unsigned pos = atomicAdd(&binFill[b], 1u);
      bucketVals[binStart[b] + pos] = x;
    }
  }
  // tail (nm not multiple of 4): handled by one thread; empty for 2^25
  if (blockIdx.x == 0 && threadIdx.x == 0) {
    for (int i = nm & ~3; i < nm; ++i) {
      const float x = m[i];
      const int b = bin_of(x);
      const unsigned pos = atomicAdd(&binFill[b], 1u);
      bucketVals[binStart[b] + pos] = x;
    }
  }
}

// ---- exact boundary-bin scan, float4-vectorized with index-masked edges ---
template <bool INCLUSIVE, bool NEED_SUM>
__device__ __forceinline__ void scan_bin(const float* __restrict__ bv,
                                         unsigned s, unsigned e, float t,
                                         unsigned& cnt, long long& sum) {
  for (unsigned j = (s & ~3u); j < e; j += 4) {
    const float4 v = *(const float4*)(bv + j);   // 16B aligned; pad covers edge
    const float xs[4] = {v.x, v.y, v.z, v.w};
    #pragma unroll
    for (int k = 0; k < 4; ++k) {
      const unsigned idx = j + (unsigned)k;
      const float x = xs[k];
      const bool pred = INCLUSIVE ? (x <= t) : (x < t);
      if (idx >= s && idx < e && pred) {
        ++cnt;
        if (NEED_SUM) sum += fixp(x);
      }
    }
  }
}

// --------------- pass 4: exact query (3 boundary-bin scans) ----------------
__global__ __launch_bounds__(256)
void k_query(const float* __restrict__ queries, int nq,
             const float* __restrict__ bucketVals,
             const unsigned* __restrict__ binStart,
             const long long* __restrict__ cumSum,
             int* __restrict__ outI, float* __restrict__ outF) {
  int i = blockIdx.x * blockDim.x + threadIdx.x;
  if (i >= nq) return;
  const float q = queries[i];
  const float tlo = q - DELTA_W;
  const float thi = q + DELTA_W;

  // lo = count(x < q - delta)  [searchsorted side='left']
  const int bl = bin_of(tlo);
  unsigned s = binStart[bl], e = binStart[bl + 1];
  unsigned cl = 0; long long sl = 0;
  scan_bin<false, true>(bucketVals, s, e, tlo, cl, sl);
  const unsigned lo = s + cl;
  const long long sumLo = cumSum[bl] + sl;

  // hi = count(x <= q + delta) [searchsorted side='right']
  const int bh = bin_of(thi);
  s = binStart[bh]; e = binStart[bh + 1];
  unsigned ch = 0; long long sh = 0;
  scan_bin<true, true>(bucketVals, s, e, thi, ch, sh);
  const unsigned hi = s + ch;
  const long long sumHi = cumSum[bh] + sh;

  // cdf_i = count(x <= q)      [searchsorted side='right'] (no sum needed)
  const int bq = bin_of(q);
  s = binStart[bq]; e = binStart[bq + 1];
  unsigned cq = 0; long long dummy = 0;
  scan_bin<true, false>(bucketVals, s, e, q, cq, dummy);
  const unsigned ci = s + cq;

  const float cnt = (float)(hi - lo);
  const float sum_win = (float)((double)(sumHi - sumLo) * INV_FIX);
  const float inv2d = 1.0f / (2.0f * DELTA_W);
  const float cdf_f = (float)lo + (cnt * (q + DELTA_W) - sum_win) * inv2d;

  outI[i] = (int)ci;
  outF[i] = cdf_f;
}

// ---------------------------------------------------------------------------
extern "C" void kernel_launch(void* const* d_in, const int* in_sizes, int n_in,
                              void* d_out, int out_size, void* d_ws, size_t ws_size,
                              hipStream_t stream) {
  const float* matrix  = (const float*)d_in[0];
  const float* queries = (const float*)d_in[1];
  const int nm = in_sizes[0];   // 1<<25
  const int nq = in_sizes[1];   // 1<<22
  (void)n_in; (void)out_size; (void)ws_size;

  // Workspace carve-out (8-byte arrays first; ~156 MB total).
  char* ws = (char*)d_ws;
  size_t off = 0;
  auto carve = [&](size_t bytes) -> void* {
    void* p = (void*)(ws + off);
    off = (off + bytes + 255) & ~(size_t)255;
    return p;
  };
  long long* binSum      = (long long*)carve(sizeof(long long) * NBINS);
  long long* cumSum      = (long long*)carve(sizeof(long long) * (NBINS + 1));
  long long* chunkSum    = (long long*)carve(sizeof(long long) * NCHUNK);
  long long* chunkOffSum = (long long*)carve(sizeof(long long) * NCHUNK);
  unsigned*  binCnt      = (unsigned*)carve(sizeof(unsigned) * NBINS);
  unsigned*  binFill     = (unsigned*)carve(sizeof(unsigned) * NBINS);
  unsigned*  binStart    = (unsigned*)carve(sizeof(unsigned) * (NBINS + 1));
  unsigned*  chunkCnt    = (unsigned*)carve(sizeof(unsigned) * NCHUNK);
  unsigned*  chunkOffCnt = (unsigned*)carve(sizeof(unsigned) * NCHUNK);
  float*     bucketVals  = (float*)carve(sizeof(float) * ((size_t)nm + 4)); // +4 pad for float4 edge reads

  k_init<<<1024, 256, 0, stream>>>(binCnt, binFill, binSum);

  const int ntile = (nm + TILE_F - 1) / TILE_F;
  const int hblk = ntile < 4096 ? ntile : 4096;
  k_hist<<<hblk, HIST_THREADS, 0, stream>>>(matrix, nm, binCnt,
                                            (unsigned long long*)binSum);

  k_reduce<<<NCHUNK, 256, 0, stream>>>(binCnt, binSum, chunkCnt, chunkSum);
  k_scanchunks<<<1, 1, 0, stream>>>(chunkCnt, chunkSum, chunkOffCnt, chunkOffSum,
                                    binStart, cumSum);
  k_scanwithin<<<NCHUNK, 256, 0, stream>>>(binCnt, binSum, chunkOffCnt,
                                           chunkOffSum, binStart, cumSum);

  const int sblk = ((nm / 4) + 255) / 256 < 4096 ? ((nm / 4) + 255) / 256 : 4096;
  k_scatter<<<sblk, 256, 0, stream>>>(matrix, nm, binStart, binFill, bucketVals);

  int*   outI = (int*)d_out;
  float* outF = ((float*)d_out) + nq;
  k_query<<<(nq + 255) / 256, 256, 0, stream>>>(queries, nq, bucketVals,
                                                binStart, cumSum, outI, outF);
}